// CommNet_84146999263282
// MI455X (gfx1250) — compile-verified
//
#include <hip/hip_runtime.h>

typedef __attribute__((ext_vector_type(2))) float v2f;
typedef __attribute__((ext_vector_type(8))) float v8f;

namespace {
constexpr int kB = 1024;   // batches
constexpr int kM = 64;     // agents per batch
constexpr int kH = 256;    // hidden
constexpr int kA = 16;     // actions
constexpr int kSteps = 4;  // comm steps
constexpr int kPitch = 260;        // kH + 4 pad floats -> conflict-free strided LDS reads
constexpr int kMT = 4;             // M tiles (64/16)
constexpr int kNTpw = 2;           // N tiles per wave (16 ntiles / 8 waves)
constexpr int kLdsFloats = 3 * kM * kPitch + 2 * kPitch;  // h, P0(h0), t, S, u
}  // namespace

__launch_bounds__(256)
__global__ void commnet_wmma_kernel(
    const int* __restrict__ ids, const float* __restrict__ emb,
    const float* __restrict__ W1, const float* __restrict__ b1,
    const float* __restrict__ W2, const float* __restrict__ b2,
    const float* __restrict__ Wd, const float* __restrict__ bd,
    float* __restrict__ out) {
  extern __shared__ float smem[];
  float* sh = smem;                 // h  : kM x kPitch (current hidden state)
  float* sp = sh + kM * kPitch;     // h0 during setup, then P0 = h0@W1c + b1
  float* st = sp + kM * kPitch;     // t  : relu(pre-activation)
  float* sS = st + kM * kPitch;     // S  : column sums of h (kH)
  float* sU = sS + kPitch;          // u  : inv * (S @ W1b)   (kH)

  const int b     = blockIdx.x;
  const int tid   = threadIdx.x;      // 256 threads = 8 wave32
  const int wave  = tid >> 5;
  const int lane  = tid & 31;
  const int lrow  = lane & 15;        // M (A frag) / N (B,D frag) index in tile
  const int khalf = (lane >> 4) << 1; // K offset of this half-wave in a K=4 step
  const int hi8   = (lane >> 4) << 3; // M offset of this half-wave in C/D tiles
  const float inv = 1.0f / (float)(kM - 1);
  const int nb    = wave * (kNTpw * 16);  // first output column of this wave

  // ---- gather: h0 = emb[agent_ids[b]], h = h0 (one coalesced row per iter) --
  for (int m = 0; m < kM; ++m) {
    const int id  = ids[b * kM + m];
    const float v = emb[id * kH + tid];
    sp[m * kPitch + tid] = v;
    sh[m * kPitch + tid] = v;
  }
  __syncthreads();

  // ==== one-time: P0 = h0 @ W1[2H:3H] + b1, written in place over h0 ========
  {
    v8f accP[kMT][kNTpw];
    const v8f vzero = {};
    for (int mt = 0; mt < kMT; ++mt)
      for (int j = 0; j < kNTpw; ++j) accP[mt][j] = vzero;
#pragma unroll 4
    for (int k0 = 0; k0 < kH; k0 += 4) {
      const int kk = k0 + khalf;
      v2f bf[kNTpw];
      for (int j = 0; j < kNTpw; ++j) {
        const int n = nb + j * 16 + lrow;
        bf[j].x = W1[(2 * kH + kk) * kH + n];
        bf[j].y = W1[(2 * kH + kk + 1) * kH + n];
      }
      for (int mt = 0; mt < kMT; ++mt) {
        const v2f a = *(const v2f*)&sp[(mt * 16 + lrow) * kPitch + kk];
        for (int j = 0; j < kNTpw; ++j)
          accP[mt][j] = __builtin_amdgcn_wmma_f32_16x16x4_f32(
              false, a, false, bf[j], (short)0, accP[mt][j], false, false);
      }
    }
    __syncthreads();  // everyone done reading h0 before it is overwritten
    for (int j = 0; j < kNTpw; ++j) {
      const int n = nb + j * 16 + lrow;
      const float bias = b1[n];
      for (int mt = 0; mt < kMT; ++mt) {
        const int mrow = mt * 16 + hi8;
#pragma unroll
        for (int r = 0; r < 8; ++r)
          sp[(mrow + r) * kPitch + n] = accP[mt][j][r] + bias;
      }
    }
    // sp tiles are wave-private from here on: each wave later reads back
    // exactly the (m, n) positions it just wrote -> no barrier needed.
  }

  for (int step = 0; step < kSteps; ++step) {
    __syncthreads();
    // ---- S[k] = sum_m h[m][k] (thread tid owns column tid) ----
    {
      float s = 0.f;
#pragma unroll 8
      for (int m = 0; m < kM; ++m) s += sh[m * kPitch + tid];
      sS[tid] = s;
    }
    __syncthreads();

    // ---- u[n] = inv * sum_k S[k] * W1[H + k][n]  (column n = tid) ----
    {
      float acc = 0.f;
#pragma unroll 4
      for (int k = 0; k < kH; ++k)
        acc = fmaf(sS[k], W1[(kH + k) * kH + tid], acc);
      sU[tid] = acc * inv;  // consumed only by this wave's epilogue lanes
    }

    // ==== GEMM1': pre = h @ (W1a - inv*W1b); t = relu(pre + P0 + u) =========
    v8f acc1[kMT][kNTpw];
    const v8f vzero = {};
    for (int mt = 0; mt < kMT; ++mt)
      for (int j = 0; j < kNTpw; ++j) acc1[mt][j] = vzero;

#pragma unroll 2
    for (int k0 = 0; k0 < kH; k0 += 4) {
      const int kk = k0 + khalf;
      v2f bf[kNTpw];
      for (int j = 0; j < kNTpw; ++j) {
        const int n = nb + j * 16 + lrow;
        const float a0 = W1[kk * kH + n];
        const float c0 = W1[(kH + kk) * kH + n];
        const float a1 = W1[(kk + 1) * kH + n];
        const float c1 = W1[(kH + kk + 1) * kH + n];
        bf[j].x = fmaf(-inv, c0, a0);  // effective weight W1a - inv*W1b
        bf[j].y = fmaf(-inv, c1, a1);
      }
      for (int mt = 0; mt < kMT; ++mt) {
        const v2f a = *(const v2f*)&sh[(mt * 16 + lrow) * kPitch + kk];
        for (int j = 0; j < kNTpw; ++j)
          acc1[mt][j] = __builtin_amdgcn_wmma_f32_16x16x4_f32(
              false, a, false, bf[j], (short)0, acc1[mt][j], false, false);
      }
    }
    for (int j = 0; j < kNTpw; ++j) {
      const int n = nb + j * 16 + lrow;
      const float u = sU[n];
      for (int mt = 0; mt < kMT; ++mt) {
        const int mrow = mt * 16 + hi8;
#pragma unroll
        for (int r = 0; r < 8; ++r) {
          const float v = acc1[mt][j][r] + sp[(mrow + r) * kPitch + n] + u;
          st[(mrow + r) * kPitch + n] = fmaxf(v, 0.f);
        }
      }
    }
    __syncthreads();

    // ==== GEMM2: h += t @ W2 + b2 ==========================================
    v8f acc2[kMT][kNTpw];
    for (int mt = 0; mt < kMT; ++mt)
      for (int j = 0; j < kNTpw; ++j) acc2[mt][j] = vzero;

#pragma unroll 4
    for (int k0 = 0; k0 < kH; k0 += 4) {
      const int kk = k0 + khalf;
      v2f bf[kNTpw];
      for (int j = 0; j < kNTpw; ++j) {
        const int n = nb + j * 16 + lrow;
        bf[j].x = W2[kk * kH + n];
        bf[j].y = W2[(kk + 1) * kH + n];
      }
      for (int mt = 0; mt < kMT; ++mt) {
        const v2f a = *(const v2f*)&st[(mt * 16 + lrow) * kPitch + kk];
        for (int j = 0; j < kNTpw; ++j)
          acc2[mt][j] = __builtin_amdgcn_wmma_f32_16x16x4_f32(
              false, a, false, bf[j], (short)0, acc2[mt][j], false, false);
      }
    }
    for (int j = 0; j < kNTpw; ++j) {
      const int n = nb + j * 16 + lrow;
      const float bias = b2[n];
      for (int mt = 0; mt < kMT; ++mt) {
        const int mrow = mt * 16 + hi8;
#pragma unroll
        for (int r = 0; r < 8; ++r)
          sh[(mrow + r) * kPitch + n] += acc2[mt][j][r] + bias;
      }
    }
  }

  __syncthreads();
  // ==== final: logits = h @ Wd + bd (N = 16 -> one ntile, waves 0..3) =======
  if (wave < kMT) {  // wave-uniform branch -> EXEC all-1 inside
    v8f accL = {};
#pragma unroll 4
    for (int k0 = 0; k0 < kH; k0 += 4) {
      const int kk = k0 + khalf;
      v2f bf;
      bf.x = Wd[kk * kA + lrow];
      bf.y = Wd[(kk + 1) * kA + lrow];
      const v2f a = *(const v2f*)&sh[(wave * 16 + lrow) * kPitch + kk];
      accL = __builtin_amdgcn_wmma_f32_16x16x4_f32(
          false, a, false, bf, (short)0, accL, false, false);
    }
    const float bias = bd[lrow];
    const int mrow = wave * 16 + hi8;
#pragma unroll
    for (int r = 0; r < 8; ++r)
      out[(b * kM + mrow + r) * kA + lrow] = accL[r] + bias;
  }
}

extern "C" void kernel_launch(void* const* d_in, const int* in_sizes, int n_in,
                              void* d_out, int out_size, void* d_ws, size_t ws_size,
                              hipStream_t stream) {
  (void)in_sizes; (void)n_in; (void)out_size; (void)d_ws; (void)ws_size;
  const int*   ids = (const int*)d_in[0];
  const float* emb = (const float*)d_in[1];
  const float* W1  = (const float*)d_in[2];
  const float* b1  = (const float*)d_in[3];
  const float* W2  = (const float*)d_in[4];
  const float* b2  = (const float*)d_in[5];
  const float* Wd  = (const float*)d_in[6];
  const float* bd  = (const float*)d_in[7];
  float* out = (float*)d_out;

  const size_t lds_bytes = (size_t)kLdsFloats * sizeof(float);  // ~202 KB < 320 KB WGP LDS
  (void)hipFuncSetAttribute((const void*)commnet_wmma_kernel,
                            hipFuncAttributeMaxDynamicSharedMemorySize,
                            (int)lds_bytes);
  commnet_wmma_kernel<<<dim3(kB), dim3(256), lds_bytes, stream>>>(
      ids, emb, W1, b1, W2, b2, Wd, bd, out);
}